// AFGCN1_4320737100470
// MI455X (gfx1250) — compile-verified
//
#include <hip/hip_runtime.h>
#include <math.h>

#define N_NODES 100000
#define N_EDGES 600000
#define F_IN 128
#define HID 32
#define NC 40
#define NCP 48          // layer-2 cols padded to 3 full 16-wide tiles
#define NB 3

typedef __attribute__((ext_vector_type(2))) float v2f;
typedef __attribute__((ext_vector_type(8))) float v8f;

// V_WMMA_F32_16X16X4_F32: D(16x16 f32) = A(16x4 f32) x B(4x16 f32) + C
static __device__ __forceinline__ v8f wmma4_f32(v2f a, v2f b, v8f c) {
    return __builtin_amdgcn_wmma_f32_16x16x4_f32(
        false, a, false, b, (short)0, c, false, false);
}

// ---------------------------------------------------------------- utilities
__global__ void __launch_bounds__(256) k_zero4(float4* __restrict__ p, int n4) {
    int i = blockIdx.x * blockDim.x + threadIdx.x;
    if (i < n4) p[i] = make_float4(0.f, 0.f, 0.f, 0.f);
}

// Sum the B=3 branch weights into WMMA-B-fragment-swizzled layout:
//   Weff1[((s*64 + p)*32 + o)*2 + j] = sum_b W1[b, s, 2p+j, o]   (K=128 -> 64 pairs)
//   Weff2[((s*16 + p)*48 + o)*2 + j] = sum_b W2[b, s, 2p+j, o]   (o>=40 zero-padded)
// so each lane's B fragment {W[kr,o], W[kr+1,o]} is one contiguous 8B load.
__global__ void __launch_bounds__(256) k_weff(
        const float* __restrict__ W1, const float* __restrict__ b1,
        const float* __restrict__ W2, const float* __restrict__ b2,
        float* __restrict__ Weff1, float* __restrict__ Weff2,
        float* __restrict__ bs1, float* __restrict__ bs2) {
    int i = blockIdx.x * blockDim.x + threadIdx.x;
    if (i < 2 * F_IN * HID) {                       // 8192 swizzled elements
        int j = i & 1;
        int t = i >> 1;
        int o = t % HID;  t /= HID;
        int p = t % (F_IN / 2);
        int s = t / (F_IN / 2);
        int f = 2 * p + j;
        float acc = 0.f;
        for (int b = 0; b < NB; ++b)
            acc += W1[(b * 2 + s) * F_IN * HID + f * HID + o];
        Weff1[i] = acc;
        return;
    }
    i -= 2 * F_IN * HID;
    if (i < 2 * (HID / 2) * NCP * 2) {              // 3072 swizzled elements
        int j = i & 1;
        int t = i >> 1;
        int o = t % NCP;  t /= NCP;
        int p = t % (HID / 2);
        int s = t / (HID / 2);
        int f = 2 * p + j;
        float acc = 0.f;
        if (o < NC)
            for (int b = 0; b < NB; ++b)
                acc += W2[(b * 2 + s) * HID * NC + f * NC + o];
        Weff2[i] = acc;
        return;
    }
    i -= 2 * (HID / 2) * NCP * 2;
    if (i < HID) {
        float s = 0.f;
        for (int b = 0; b < NB; ++b) s += b1[b * HID + i];
        bs1[i] = s;
        return;
    }
    i -= HID;
    if (i < NC) {
        float s = 0.f;
        for (int b = 0; b < NB; ++b) s += b2[b * NC + i];
        bs2[i] = s;
    }
}

// deg[row[e]] += 1  (float segment_sum of ones)
__global__ void __launch_bounds__(256) k_deg(const int* __restrict__ row,
                                             float* __restrict__ deg) {
    int e = blockIdx.x * blockDim.x + threadIdx.x;
    if (e < N_EDGES) atomicAdd(&deg[row[e]], 1.0f);
}

// norm[e] = -dis(deg[row]) * dis(deg[col]),  dis(d) = d>0 ? rsqrt(max(d,1)) : 0
__global__ void __launch_bounds__(256) k_norm(const int* __restrict__ row,
                                              const int* __restrict__ col,
                                              const float* __restrict__ deg,
                                              float* __restrict__ norm) {
    int e = blockIdx.x * blockDim.x + threadIdx.x;
    if (e >= N_EDGES) return;
    float dr = deg[row[e]], dc = deg[col[e]];
    float a = (dr > 0.f) ? rsqrtf(fmaxf(dr, 1.f)) : 0.f;
    float b = (dc > 0.f) ? rsqrtf(fmaxf(dc, 1.f)) : 0.f;
    norm[e] = -a * b;
}

// dst[col[e], :] += src[row[e], :] * norm[e]; one thread per (edge, 4-feature chunk)
__global__ void __launch_bounds__(256) k_spmm(
        const float* __restrict__ src, const int* __restrict__ row,
        const int* __restrict__ col, const float* __restrict__ norm,
        float* __restrict__ dst, int shift, int total) {
    int t = blockIdx.x * blockDim.x + threadIdx.x;
    if (t >= total) return;
    int e = t >> shift;
    int c = t & ((1 << shift) - 1);
    int F = 4 << shift;
    float nv = norm[e];
    const float4 v = *(const float4*)(src + (size_t)row[e] * F + c * 4);
    float* d = dst + (size_t)col[e] * F + c * 4;
    atomicAdd(d + 0, v.x * nv);
    atomicAdd(d + 1, v.y * nv);
    atomicAdd(d + 2, v.z * nv);
    atomicAdd(d + 3, v.w * nv);
}

// h1 = relu(x @ Weff1[0] + tx1 @ Weff1[1] + bs1); one wave per 16 rows, 2 col tiles
__global__ void __launch_bounds__(256) k_gemm1(
        const float* __restrict__ x, const float* __restrict__ tx1,
        const float* __restrict__ Weff1, const float* __restrict__ bs1,
        float* __restrict__ h1) {
    int gwave = (blockIdx.x * blockDim.x + threadIdx.x) >> 5;
    int lane = threadIdx.x & 31;
    int row0 = gwave * 16;
    if (row0 >= N_NODES) return;
    int mrow = lane & 15;
    int khalf = lane >> 4;                  // 0: K = k0,k0+1 ; 1: K = k0+2,k0+3
    const float* W0 = Weff1;                          // k=0 slab (swizzled)
    const float* W1p = Weff1 + (F_IN / 2) * HID * 2;  // k=1 slab
    const float* xr = x + (size_t)(row0 + mrow) * F_IN;
    const float* tr = tx1 + (size_t)(row0 + mrow) * F_IN;
    v8f acc0 = {}, acc1 = {};
    for (int k = 0; k < F_IN; k += 4) {
        int kp = (k >> 1) + khalf;          // K-pair index
        v2f a = *(const v2f*)(xr + k + khalf * 2);
        v2f b0 = *(const v2f*)(W0 + ((size_t)kp * HID + mrow) * 2);
        v2f b1 = *(const v2f*)(W0 + ((size_t)kp * HID + 16 + mrow) * 2);
        acc0 = wmma4_f32(a, b0, acc0);
        acc1 = wmma4_f32(a, b1, acc1);
    }
    for (int k = 0; k < F_IN; k += 4) {
        int kp = (k >> 1) + khalf;
        v2f a = *(const v2f*)(tr + k + khalf * 2);
        v2f b0 = *(const v2f*)(W1p + ((size_t)kp * HID + mrow) * 2);
        v2f b1 = *(const v2f*)(W1p + ((size_t)kp * HID + 16 + mrow) * 2);
        acc0 = wmma4_f32(a, b0, acc0);
        acc1 = wmma4_f32(a, b1, acc1);
    }
    float bb0 = bs1[mrow], bb1 = bs1[16 + mrow];
    for (int r = 0; r < 8; ++r) {
        int rw = row0 + r + 8 * khalf;       // C layout: lanes>=16 hold M=r+8
        float* o = h1 + (size_t)rw * HID;
        o[mrow] = fmaxf(acc0[r] + bb0, 0.f);
        o[16 + mrow] = fmaxf(acc1[r] + bb1, 0.f);
    }
}

// logits = h1 @ Weff2[0] + tx2 @ Weff2[1] + bs2; 48 padded cols, store 40
__global__ void __launch_bounds__(256) k_gemm2(
        const float* __restrict__ h1, const float* __restrict__ tx2,
        const float* __restrict__ Weff2, const float* __restrict__ bs2,
        float* __restrict__ logits) {
    int gwave = (blockIdx.x * blockDim.x + threadIdx.x) >> 5;
    int lane = threadIdx.x & 31;
    int row0 = gwave * 16;
    if (row0 >= N_NODES) return;
    int mrow = lane & 15;
    int khalf = lane >> 4;
    const float* W0 = Weff2;                          // swizzled, 48-col padded
    const float* W1p = Weff2 + (HID / 2) * NCP * 2;
    const float* hr = h1 + (size_t)(row0 + mrow) * HID;
    const float* tr = tx2 + (size_t)(row0 + mrow) * HID;
    v8f acc0 = {}, acc1 = {}, acc2 = {};
    for (int k = 0; k < HID; k += 4) {
        int kp = (k >> 1) + khalf;
        v2f a = *(const v2f*)(hr + k + khalf * 2);
        v2f a2 = *(const v2f*)(tr + k + khalf * 2);
        v2f b0 = *(const v2f*)(W0 + ((size_t)kp * NCP + mrow) * 2);
        v2f b1 = *(const v2f*)(W0 + ((size_t)kp * NCP + 16 + mrow) * 2);
        v2f b2 = *(const v2f*)(W0 + ((size_t)kp * NCP + 32 + mrow) * 2);
        acc0 = wmma4_f32(a, b0, acc0);
        acc1 = wmma4_f32(a, b1, acc1);
        acc2 = wmma4_f32(a, b2, acc2);
        b0 = *(const v2f*)(W1p + ((size_t)kp * NCP + mrow) * 2);
        b1 = *(const v2f*)(W1p + ((size_t)kp * NCP + 16 + mrow) * 2);
        b2 = *(const v2f*)(W1p + ((size_t)kp * NCP + 32 + mrow) * 2);
        acc0 = wmma4_f32(a2, b0, acc0);
        acc1 = wmma4_f32(a2, b1, acc1);
        acc2 = wmma4_f32(a2, b2, acc2);
    }
    bool c2ok = (mrow < NC - 32);
    float bb0 = bs2[mrow], bb1 = bs2[16 + mrow];
    float bb2 = c2ok ? bs2[32 + mrow] : 0.f;
    for (int r = 0; r < 8; ++r) {
        int rw = row0 + r + 8 * khalf;
        float* o = logits + (size_t)rw * NC;
        o[mrow] = acc0[r] + bb0;
        o[16 + mrow] = acc1[r] + bb1;
        if (c2ok) o[32 + mrow] = acc2[r] + bb2;
    }
}

// row-wise log_softmax over C=40, one wave32 per row
__global__ void __launch_bounds__(256) k_lsm(const float* __restrict__ logits,
                                             float* __restrict__ out) {
    int gwave = (blockIdx.x * blockDim.x + threadIdx.x) >> 5;
    int lane = threadIdx.x & 31;
    if (gwave >= N_NODES) return;
    const float* rowp = logits + (size_t)gwave * NC;
    float a = rowp[lane];                               // lane < 32 < 40
    float b = (lane < NC - 32) ? rowp[32 + lane] : -INFINITY;
    float m = fmaxf(a, b);
    for (int off = 16; off > 0; off >>= 1)
        m = fmaxf(m, __shfl_xor(m, off, 32));
    float s = expf(a - m) + ((lane < NC - 32) ? expf(b - m) : 0.f);
    for (int off = 16; off > 0; off >>= 1)
        s += __shfl_xor(s, off, 32);
    float ls = logf(s);
    float* o = out + (size_t)gwave * NC;
    o[lane] = a - m - ls;
    if (lane < NC - 32) o[32 + lane] = b - m - ls;
}

extern "C" void kernel_launch(void* const* d_in, const int* in_sizes, int n_in,
                              void* d_out, int out_size, void* d_ws, size_t ws_size,
                              hipStream_t stream) {
    const float* x = (const float*)d_in[0];
    const int* row = (const int*)d_in[1];
    const int* col = ((const int*)d_in[1]) + N_EDGES;
    const float* W1 = (const float*)d_in[2];
    const float* b1 = (const float*)d_in[3];
    const float* W2 = (const float*)d_in[4];
    const float* b2 = (const float*)d_in[5];
    float* out = (float*)d_out;

    // Workspace layout (floats):
    //  [buf1: N*128]  tx1, later reused as {tx2: N*32 | logits: N*40}
    //  [h1:   N*32 ]
    //  [deg:  N    ]
    //  [norm: E    ]
    //  [Weff1 8192][Weff2 3072][bs1 32][bs2 40]
    float* ws = (float*)d_ws;
    float* buf1 = ws;
    float* h1 = buf1 + (size_t)N_NODES * F_IN;
    float* deg = h1 + (size_t)N_NODES * HID;
    float* norm = deg + N_NODES;
    float* Weff1 = norm + N_EDGES;
    float* Weff2 = Weff1 + 2 * F_IN * HID;
    float* bs1 = Weff2 + 2 * (HID / 2) * NCP * 2;
    float* bs2 = bs1 + HID;

    float* tx1 = buf1;
    float* tx2 = buf1;                       // reuse after gemm1
    float* logits = buf1 + (size_t)N_NODES * HID;

    const int TB = 256;
    // zero contiguous [buf1 | h1 | deg] = N*(128+32+1) accumulators (float4)
    int nz4 = N_NODES * (F_IN + HID + 1) / 4;
    k_zero4<<<(nz4 + TB - 1) / TB, TB, 0, stream>>>((float4*)ws, nz4);

    int nweff = 2 * F_IN * HID + 2 * (HID / 2) * NCP * 2 + HID + NC;
    k_weff<<<(nweff + TB - 1) / TB, TB, 0, stream>>>(W1, b1, W2, b2,
                                                     Weff1, Weff2, bs1, bs2);

    k_deg<<<(N_EDGES + TB - 1) / TB, TB, 0, stream>>>(row, deg);
    k_norm<<<(N_EDGES + TB - 1) / TB, TB, 0, stream>>>(row, col, deg, norm);

    int sp1 = N_EDGES * (F_IN / 4);          // shift=5 -> 32 chunks/edge
    k_spmm<<<(sp1 + TB - 1) / TB, TB, 0, stream>>>(x, row, col, norm, tx1, 5, sp1);

    int nwaves1 = (N_NODES + 15) / 16;       // one wave per 16 rows
    int nthr1 = nwaves1 * 32;
    k_gemm1<<<(nthr1 + TB - 1) / TB, TB, 0, stream>>>(x, tx1, Weff1, bs1, h1);

    // zero tx2 region before scatter-add
    int nz24 = N_NODES * HID / 4;
    k_zero4<<<(nz24 + TB - 1) / TB, TB, 0, stream>>>((float4*)tx2, nz24);

    int sp2 = N_EDGES * (HID / 4);           // shift=3 -> 8 chunks/edge
    k_spmm<<<(sp2 + TB - 1) / TB, TB, 0, stream>>>(h1, row, col, norm, tx2, 3, sp2);

    k_gemm2<<<(nthr1 + TB - 1) / TB, TB, 0, stream>>>(h1, tx2, Weff2, bs2, logits);

    int nthr_lsm = N_NODES * 32;             // one wave per row
    k_lsm<<<(nthr_lsm + TB - 1) / TB, TB, 0, stream>>>(logits, out);
}